// RoIHeadTemplate_11536282157120
// MI455X (gfx1250) — compile-verified
//
#include <hip/hip_runtime.h>
#include <hip/hip_bf16.h>
#include <stdint.h>

#define BATCH 4
#define NPTS  32768
#define NCLS  3
#define PRE   4096
#define POST  512
#define MASKW (PRE / 32)   // 128 words per row
#define THRESH 0.7f

typedef float v2f __attribute__((ext_vector_type(2)));
typedef float v8f __attribute__((ext_vector_type(8)));

// ---------------------------------------------------------------- K1: max / argmax over classes
__global__ void k_score_label(const float* __restrict__ cls,
                              float* __restrict__ scores, int* __restrict__ labels) {
    int t = blockIdx.x * blockDim.x + threadIdx.x;
    if (t >= BATCH * NPTS) return;
    const float* p = cls + (size_t)t * NCLS;
    float best = p[0];
    int bl = 0;
#pragma unroll
    for (int c = 1; c < NCLS; ++c) {
        float v = p[c];
        if (v > best) { best = v; bl = c; }   // strict > == first-occurrence argmax
    }
    scores[t] = best;
    labels[t] = bl;
}

// ---------------------------------------------------------------- K2: full bitonic sort in LDS (top-4096)
// One workgroup per batch item. 32768 x (key u32 + idx u32) = 256 KB LDS,
// legal on CDNA5 (320 KB per workgroup). Key = order-preserving float map,
// sorted descending with ascending-index tie-break (matches lax.top_k).
__global__ __launch_bounds__(1024) void k_topk_sort(const float* __restrict__ scores,
                                                    int* __restrict__ topidx,
                                                    float* __restrict__ topscore) {
    __shared__ unsigned int sm[NPTS];
    __shared__ unsigned int sid[NPTS];
    int b = blockIdx.x;
    const float* sc = scores + (size_t)b * NPTS;

    for (int i = threadIdx.x; i < NPTS; i += blockDim.x) {
        unsigned u = __float_as_uint(sc[i]);
        unsigned m = (u & 0x80000000u) ? ~u : (u | 0x80000000u);  // monotone map
        sm[i] = m;
        sid[i] = (unsigned)i;
    }
    __syncthreads();

    for (int k = 2; k <= NPTS; k <<= 1) {
        for (int j = k >> 1; j > 0; j >>= 1) {
            for (int i = threadIdx.x; i < NPTS; i += blockDim.x) {
                int l = i ^ j;
                if (l > i) {
                    bool up = ((i & k) == 0);
                    unsigned mi = sm[i], ml = sm[l];
                    unsigned ii = sid[i], il = sid[l];
                    // "iBeforeL": i should precede l in final order (higher score, tie lower idx)
                    bool iBeforeL = (mi > ml) || (mi == ml && ii < il);
                    bool doSwap = up ? (!iBeforeL) : iBeforeL;
                    if (doSwap) {
                        sm[i] = ml; sm[l] = mi;
                        sid[i] = il; sid[l] = ii;
                    }
                }
            }
            __syncthreads();
        }
    }

    for (int t = threadIdx.x; t < PRE; t += blockDim.x) {
        int id = (int)sid[t];
        topidx[b * PRE + t] = id;
        topscore[b * PRE + t] = sc[id];
    }
}

// ---------------------------------------------------------------- K3: box SoA prep for top-4096
__global__ void k_boxprep(const float* __restrict__ boxes, const int* __restrict__ topidx,
                          float* __restrict__ x1, float* __restrict__ x2,
                          float* __restrict__ y1, float* __restrict__ y2,
                          float* __restrict__ area) {
    int t = blockIdx.x * blockDim.x + threadIdx.x;
    if (t >= BATCH * PRE) return;
    int b = t / PRE;
    int id = topidx[t];
    const float* bx = boxes + ((size_t)b * NPTS + id) * 7;
    float x = bx[0], y = bx[1], dx = bx[3], dy = bx[4];
    x1[t] = x - dx * 0.5f;
    x2[t] = x + dx * 0.5f;
    y1[t] = y - dy * 0.5f;
    y2[t] = y + dy * 0.5f;
    area[t] = dx * dy;
}

// ---------------------------------------------------------------- K4: IoU suppression bitmask via WMMA
// One wave per 16-row x 32-col strip. Union base (area_i + area_j) is a rank-2
// outer product computed with V_WMMA_F32_16X16X4_F32:
//   A[m][0]=area_row[m], A[m][1]=1 ; B[0][n]=1, B[1][n]=area_col[n]  => D=area_i+area_j
// Intersection via VALU; row-major mask bits assembled with __ballot (bits 0-15
// = row k, bits 16-31 = row k+8 per the 16x16 f32 C/D lane layout).
__global__ __launch_bounds__(32) void k_iou_mask(const float* __restrict__ X1,
                                                 const float* __restrict__ X2,
                                                 const float* __restrict__ Y1,
                                                 const float* __restrict__ Y2,
                                                 const float* __restrict__ AREA,
                                                 unsigned int* __restrict__ mask) {
    int b = blockIdx.z;
    int rowbase = blockIdx.y * 16;
    int colbase = blockIdx.x * 32;
    int lane = threadIdx.x;
    int half = lane >> 4;        // lanes 0-15: rows k, lanes 16-31: rows k+8
    int nloc = lane & 15;        // column within 16-wide tile
    size_t base = (size_t)b * PRE;
    const float* x1 = X1 + base; const float* x2 = X2 + base;
    const float* y1 = Y1 + base; const float* y2 = Y2 + base;
    const float* ar = AREA + base;

    // Build WMMA operands (loads unconditional with clamped index; zeros via select
    // so EXEC stays all-ones around the WMMA).
    float rArea  = ar[rowbase + nloc];
    float cArea0 = ar[colbase + nloc];
    float cArea1 = ar[colbase + 16 + nloc];
    bool lo = (lane < 16);
    v2f a;  a.x  = lo ? rArea  : 0.0f;  a.y  = lo ? 1.0f   : 0.0f;  // K=0: area_i, K=1: 1
    v2f b0; b0.x = lo ? 1.0f   : 0.0f;  b0.y = lo ? cArea0 : 0.0f;  // K=0: 1, K=1: area_j
    v2f b1; b1.x = lo ? 1.0f   : 0.0f;  b1.y = lo ? cArea1 : 0.0f;
    v8f cz = {};
    v8f d0 = __builtin_amdgcn_wmma_f32_16x16x4_f32(false, a, false, b0, (short)0, cz, false, false);
    v8f d1 = __builtin_amdgcn_wmma_f32_16x16x4_f32(false, a, false, b1, (short)0, cz, false, false);

    int g0 = colbase + nloc;         // this lane's column, tile 0
    int g1 = colbase + 16 + nloc;    // this lane's column, tile 1
    float c0x1 = x1[g0], c0x2 = x2[g0], c0y1 = y1[g0], c0y2 = y2[g0];
    float c1x1 = x1[g1], c1x2 = x2[g1], c1y1 = y1[g1], c1y2 = y2[g1];

    unsigned rw[16];
#pragma unroll
    for (int k = 0; k < 8; ++k) {
        int m = rowbase + half * 8 + k;  // this lane's row for accumulator VGPR k
        float rx1 = x1[m], rx2 = x2[m], ry1 = y1[m], ry2 = y2[m];

        float ix0 = fmaxf(fminf(rx2, c0x2) - fmaxf(rx1, c0x1), 0.0f);
        float iy0 = fmaxf(fminf(ry2, c0y2) - fmaxf(ry1, c0y1), 0.0f);
        float in0 = ix0 * iy0;
        float un0 = fmaxf(d0[k] - in0, 1e-6f);
        bool sup0 = (in0 > THRESH * un0) && (g0 > m);   // iou > 0.7 and j > i
        unsigned bA = (unsigned)__ballot(sup0);

        float ix1 = fmaxf(fminf(rx2, c1x2) - fmaxf(rx1, c1x1), 0.0f);
        float iy1 = fmaxf(fminf(ry2, c1y2) - fmaxf(ry1, c1y1), 0.0f);
        float in1 = ix1 * iy1;
        float un1 = fmaxf(d1[k] - in1, 1e-6f);
        bool sup1 = (in1 > THRESH * un1) && (g1 > m);
        unsigned bB = (unsigned)__ballot(sup1);

        rw[k]     = (bA & 0xFFFFu) | (bB << 16);          // row k, cols [colbase, colbase+32)
        rw[k + 8] = (bA >> 16) | (bB & 0xFFFF0000u);      // row k+8
    }

    if (lane == 0) {
        unsigned wsel = (unsigned)(colbase >> 5);
#pragma unroll
        for (int r = 0; r < 16; ++r)
            mask[((size_t)b * PRE + rowbase + r) * MASKW + wsel] = rw[r];
    }
}

// ---------------------------------------------------------------- K5: serial NMS scan, one wave32 per batch
// Removal bitmask (4096 bits = 128 words) lives in 4 VGPRs per lane; flag check
// via __shfl (wave-uniform), mask rows streamed from L2. Emits packed keep words.
__global__ __launch_bounds__(32) void k_nms_scan(const unsigned int* __restrict__ mask,
                                                 unsigned int* __restrict__ keepw) {
    int b = blockIdx.x;
    int lane = threadIdx.x;
    const unsigned int* M = mask + (size_t)b * PRE * MASKW;
    unsigned rem0 = 0, rem1 = 0, rem2 = 0, rem3 = 0;  // words: lane, lane+32, lane+64, lane+96
    unsigned kw = 0;
    for (int i = 0; i < PRE; ++i) {
        int w = i >> 5;
        int src = w & 31;
        unsigned word;
        switch (w >> 5) {
            case 0:  word = (unsigned)__shfl((int)rem0, src); break;
            case 1:  word = (unsigned)__shfl((int)rem1, src); break;
            case 2:  word = (unsigned)__shfl((int)rem2, src); break;
            default: word = (unsigned)__shfl((int)rem3, src); break;
        }
        bool removed = (word >> (i & 31)) & 1u;   // uniform across wave
        if (!removed) {
            const unsigned int* row = M + (size_t)i * MASKW;
            rem0 |= row[lane];
            rem1 |= row[lane + 32];
            rem2 |= row[lane + 64];
            rem3 |= row[lane + 96];
        }
        kw |= (removed ? 0u : 1u) << (i & 31);
        if ((i & 31) == 31) {
            if (lane == 0) keepw[b * MASKW + (i >> 5)] = kw;
            kw = 0;
        }
    }
}

// ---------------------------------------------------------------- K6: stable rank-scatter finalize
// Reproduces order = argsort(where(keep, ar, PRE)) (stable): kept positions in
// order get ranks 0..Ktot-1, suppressed positions get Ktot + (order among
// suppressed). Slots r < 512 gather rois/scores/labels.
__global__ __launch_bounds__(256) void k_finalize(const unsigned int* __restrict__ keepw,
                                                  const int* __restrict__ topidx,
                                                  const float* __restrict__ topscore,
                                                  const float* __restrict__ boxes,
                                                  const int* __restrict__ labels,
                                                  float* __restrict__ out) {
    __shared__ unsigned int swords[MASKW];
    __shared__ unsigned int wpref[MASKW + 1];
    int b = blockIdx.x;
    const unsigned int* kwp = keepw + b * MASKW;
    for (int w = threadIdx.x; w < MASKW; w += blockDim.x) swords[w] = kwp[w];
    __syncthreads();
    if (threadIdx.x == 0) {
        unsigned s = 0;
        for (int w = 0; w < MASKW; ++w) { wpref[w] = s; s += __popc(swords[w]); }
        wpref[MASKW] = s;
    }
    __syncthreads();
    unsigned Ktot = wpref[MASKW];

    float* rois    = out;                                   // [B][POST][7]
    float* oscores = out + (size_t)BATCH * POST * 7;        // [B][POST]
    float* olabels = oscores + (size_t)BATCH * POST;        // [B][POST]

    for (int p = threadIdx.x; p < PRE; p += blockDim.x) {
        int w = p >> 5, bit = p & 31;
        unsigned word = swords[w];
        unsigned keptBefore = wpref[w] + __popc(word & ((1u << bit) - 1u));
        bool kept = (word >> bit) & 1u;
        unsigned r = kept ? keptBefore : (Ktot + (unsigned)p - keptBefore);
        if (r < POST) {
            int id = topidx[b * PRE + p];
            float valid = kept ? 1.0f : 0.0f;
            const float* src = boxes + ((size_t)b * NPTS + id) * 7;
            float* dst = rois + ((size_t)b * POST + r) * 7;
#pragma unroll
            for (int q = 0; q < 7; ++q) dst[q] = src[q] * valid;
            oscores[b * POST + r] = topscore[b * PRE + p] * valid;
            int lab = labels[b * NPTS + id];
            olabels[b * POST + r] = kept ? (float)(lab + 1) : 1.0f;
        }
    }
}

// ---------------------------------------------------------------- launch
extern "C" void kernel_launch(void* const* d_in, const int* in_sizes, int n_in,
                              void* d_out, int out_size, void* d_ws, size_t ws_size,
                              hipStream_t stream) {
    (void)in_sizes; (void)n_in; (void)out_size; (void)ws_size;
    const float* cls   = (const float*)d_in[0];   // [B][N][3]
    const float* boxes = (const float*)d_in[1];   // [B][N][7]
    float* out = (float*)d_out;

    char* ws = (char*)d_ws;
    size_t off = 0;
    auto alloc = [&](size_t bytes) -> void* {
        void* p = ws + off;
        off = (off + bytes + 255) & ~(size_t)255;
        return p;
    };
    float*    scores   = (float*)alloc((size_t)BATCH * NPTS * 4);
    int*      labels   = (int*)  alloc((size_t)BATCH * NPTS * 4);
    int*      topidx   = (int*)  alloc((size_t)BATCH * PRE * 4);
    float*    topscore = (float*)alloc((size_t)BATCH * PRE * 4);
    float*    x1       = (float*)alloc((size_t)BATCH * PRE * 4);
    float*    x2       = (float*)alloc((size_t)BATCH * PRE * 4);
    float*    y1       = (float*)alloc((size_t)BATCH * PRE * 4);
    float*    y2       = (float*)alloc((size_t)BATCH * PRE * 4);
    float*    area     = (float*)alloc((size_t)BATCH * PRE * 4);
    unsigned* mask     = (unsigned*)alloc((size_t)BATCH * PRE * MASKW * 4);  // 8 MB
    unsigned* keepw    = (unsigned*)alloc((size_t)BATCH * MASKW * 4);

    k_score_label<<<(BATCH * NPTS + 255) / 256, 256, 0, stream>>>(cls, scores, labels);
    k_topk_sort<<<BATCH, 1024, 0, stream>>>(scores, topidx, topscore);
    k_boxprep<<<(BATCH * PRE + 255) / 256, 256, 0, stream>>>(boxes, topidx, x1, x2, y1, y2, area);
    k_iou_mask<<<dim3(PRE / 32, PRE / 16, BATCH), 32, 0, stream>>>(x1, x2, y1, y2, area, mask);
    k_nms_scan<<<BATCH, 32, 0, stream>>>(mask, keepw);
    k_finalize<<<BATCH, 256, 0, stream>>>(keepw, topidx, topscore, boxes, labels, out);
}